// HuMPS_87290915324560
// MI455X (gfx1250) — compile-verified
//
#include <hip/hip_runtime.h>

typedef __attribute__((ext_vector_type(2))) float v2f;
typedef __attribute__((ext_vector_type(8))) float v8f;

// ---------------------------------------------------------------------------
// Kernel 1: compose the two gate layers per bond with WMMA, then project.
//   Ueff_k = U1_k(4x4) @ U0_k(4x4);  M[k][x][i][j] = Ueff_k[2x][2i+j]
// Pack 4 sites per V_WMMA_F32_16X16X4_F32:
//   A(16x4) = vertical stack of the 4 sites' U1, B(4x16) = horizontal stack
//   of the same sites' U0 -> diagonal 4x4 blocks of D are the products.
// A layout (ISA 7.12.2, 32-bit A 16x4): lane l<16 holds row l, K={0,1};
// lane l>=16 holds row l-16, K={2,3}.  B mirrored: lanes 0-15 rows K={0,1},
// lanes 16-31 rows K={2,3}, col = lane&15.
// D layout: VGPR v, lanes 0-15 -> row v, col lane; lanes 16-31 -> row 8+v.
// ---------------------------------------------------------------------------
__global__ __launch_bounds__(32) void humps_prep(const float* __restrict__ U,
                                                 float* __restrict__ M) {
  const int lane = threadIdx.x;   // exactly one wave32, EXEC all ones
  const int r    = lane & 15;     // A row index == B col index
  const int half = lane >> 4;     // 0: K={0,1}   1: K={2,3}
  const int kb   = half * 2;
  const int si   = r >> 2;        // site within group of 4
  const int rc   = r & 3;         // row within U1 block == col within U0 block

  for (int g = 0; g < 16; ++g) {
    const int  site  = g * 4 + si;
    const bool valid = site < 63;
    const int  s     = valid ? site : 0;
    const float* U0  = U + s * 16;          // layer 0 gate (4x4 row-major)
    const float* U1  = U + 1008 + s * 16;   // layer 1 gate

    v2f a, b;
    a.x = valid ? U1[rc * 4 + kb]       : 0.f;
    a.y = valid ? U1[rc * 4 + kb + 1]   : 0.f;
    b.x = valid ? U0[kb * 4 + rc]       : 0.f;
    b.y = valid ? U0[(kb + 1) * 4 + rc] : 0.f;

    v8f cz = {};
    v8f d = __builtin_amdgcn_wmma_f32_16x16x4_f32(false, a, false, b,
                                                  (short)0, cz, false, false);

    // Diagonal block si lives in rows 4*si..4*si+3:
    //   rows 0-7 are held by lanes 0-15 (half 0), rows 8-15 by lanes 16-31.
    const bool mine = (half == 0) ? (si < 2) : (si >= 2);
    if (mine && valid) {
      const bool hi = (si & 1) != 0;       // row 4*si -> VGPR 0 or 4
      const float e0 = hi ? d[4] : d[0];   // Ueff row 2*0 of this block
      const float e1 = hi ? d[6] : d[2];   // Ueff row 2*1 of this block
      M[site * 8 + 0 * 4 + rc] = e0;       // M[k][0][i][j], rc = 2i+j
      M[site * 8 + 1 * 4 + rc] = e1;       // M[k][1][i][j]
    }
  }
}

// ---------------------------------------------------------------------------
// Kernel 2: bandwidth-bound streaming chain.  One thread per batch element:
// carry (2 fp32 regs) updated through 63 steps; X streamed with non-temporal
// coalesced b64 loads, next site prefetched one step ahead; M broadcast from
// LDS (all lanes read the same address).
// ---------------------------------------------------------------------------
__global__ __launch_bounds__(256) void humps_chain(const float* __restrict__ X,
                                                   const float* __restrict__ Mg,
                                                   float* __restrict__ out,
                                                   int B) {
  __shared__ float sM[504];                       // 63 * 8 floats
  for (int t = threadIdx.x; t < 504; t += 256) sM[t] = Mg[t];
  __syncthreads();

  const long b = (long)blockIdx.x * 256 + threadIdx.x;
  if (b >= B) return;

  const v2f* Xb = (const v2f*)X + b;              // site stride = B float2s
  v2f c = __builtin_nontemporal_load(Xb);         // carry = X[0][b]
  v2f x = __builtin_nontemporal_load(Xb + (long)B);
  float c0 = c.x, c1 = c.y;

#pragma unroll 7
  for (int k = 0; k < 63; ++k) {
    v2f xn = (k < 62) ? __builtin_nontemporal_load(Xb + (long)(k + 2) * B)
                      : (v2f)0.f;
    const float* Mk = &sM[k * 8];
    const float t00 = c0 * x.x, t01 = c0 * x.y;
    const float t10 = c1 * x.x, t11 = c1 * x.y;
    const float n0 = __builtin_fmaf(t00, Mk[0], __builtin_fmaf(t01, Mk[1],
                     __builtin_fmaf(t10, Mk[2], t11 * Mk[3])));
    const float n1 = __builtin_fmaf(t00, Mk[4], __builtin_fmaf(t01, Mk[5],
                     __builtin_fmaf(t10, Mk[6], t11 * Mk[7])));
    c0 = n0; c1 = n1; x = xn;
  }

  v2f res; res.x = c0 * c0; res.y = c1 * c1;
  __builtin_nontemporal_store(res, (v2f*)out + b);
}

// ---------------------------------------------------------------------------
extern "C" void kernel_launch(void* const* d_in, const int* in_sizes, int n_in,
                              void* d_out, int out_size, void* d_ws, size_t ws_size,
                              hipStream_t stream) {
  const float* X  = (const float*)d_in[0];   // (64, B, 2) fp32
  const float* U  = (const float*)d_in[1];   // (2, 63, 2,2,2,2) fp32
  float*       M  = (float*)d_ws;            // 63*8 fp32 scratch
  float*       out = (float*)d_out;          // (B, 2) fp32

  const int B = in_sizes[0] / (64 * 2);      // N=64, D=2

  humps_prep<<<1, 32, 0, stream>>>(U, M);
  humps_chain<<<(B + 255) / 256, 256, 0, stream>>>(X, M, out, B);
}